// MoEAllGatherTokenDispatcher_22162031247684
// MI455X (gfx1250) — compile-verified
//
#include <hip/hip_runtime.h>

// MoE allgather token dispatcher (ep=etp=1): permute + probs-weighted
// scatter-add round-trips to: out[t,:] = hs[t,:] * rowsum(probs[t,:]).
// probs already carries zeros at non-routed experts, so routing_map is unused.
//
// Bandwidth-bound: 132 MiB of traffic (fits in the 192 MB L2 across graph
// replays). Main stream uses b128 loads/stores. The per-token row-sum over
// 64 experts runs on the matrix pipe: V_WMMA_F32_16X16X4_F32 with B = ones
// computes D[i,j] = sum_k A[i,k], giving 16 token sums per wave in f32.

typedef __attribute__((ext_vector_type(2))) float v2f;
typedef __attribute__((ext_vector_type(8))) float v8f;

#define HDIM        1024
#define NEXP        64
#define TOK_PER_BLK 16
#define BLK_THREADS 256

__global__ __launch_bounds__(BLK_THREADS)
void moe_combine_wmma_kernel(const float* __restrict__ hs,
                             const float* __restrict__ probs,
                             float* __restrict__ out)
{
    __shared__ float s_sum[TOK_PER_BLK];

    const int t0  = blockIdx.x * TOK_PER_BLK;
    const int tid = threadIdx.x;

    // ---- Wave 0: row-sum of probs[t0:t0+16, 0:64] via chained WMMA ----
    // A layout (16x4 f32, ISA 7.12.2): lanes 0-15 hold row M=lane, K={0,1};
    // lanes 16-31 hold row M=lane-16, K={2,3}.
    // D layout: VGPR r = row r (lanes 0-15) / row r+8 (lanes 16-31),
    // replicated across columns since B is all-ones.
    if (tid < 32) {
        const int lane = tid;
        const int row  = lane & 15;
        const int kb   = (lane >> 4) << 1;   // 0 for lanes 0-15, 2 for 16-31
        const float* __restrict__ prow =
            probs + (size_t)(t0 + row) * NEXP + kb;

        v8f c = {};
        v2f b;
        b.x = 1.0f;
        b.y = 1.0f;

#pragma unroll
        for (int ek = 0; ek < NEXP; ek += 4) {
            v2f a;
            a.x = prow[ek];
            a.y = prow[ek + 1];
            // 8 args: (neg_a, A, neg_b, B, c_mod, C, reuse_a, reuse_b)
            c = __builtin_amdgcn_wmma_f32_16x16x4_f32(
                    false, a, false, b, (short)0, c, false, false);
        }

        if (lane == 0) {
#pragma unroll
            for (int r = 0; r < 8; ++r) s_sum[r] = c[r];          // tokens 0..7
        } else if (lane == 16) {
#pragma unroll
            for (int r = 0; r < 8; ++r) s_sum[8 + r] = c[r];      // tokens 8..15
        }
    }
    __syncthreads();

    // ---- All 8 waves: stream 16 tokens x 4KB with b128 ----
    // HDIM/4 == 256 float4 per token == blockDim, so iteration i covers
    // token t0+i with one fully-coalesced 4KB load + 4KB store.
    const float4* __restrict__ src = (const float4*)(hs  + (size_t)t0 * HDIM);
    float4*       __restrict__ dst = (float4*)      (out + (size_t)t0 * HDIM);

#pragma unroll 4
    for (int i = 0; i < TOK_PER_BLK; ++i) {
        const float s   = s_sum[i];
        const int   idx = i * (HDIM / 4) + tid;
        float4 v = src[idx];
        v.x *= s; v.y *= s; v.z *= s; v.w *= s;
        dst[idx] = v;
    }
}

extern "C" void kernel_launch(void* const* d_in, const int* in_sizes, int n_in,
                              void* d_out, int out_size, void* d_ws, size_t ws_size,
                              hipStream_t stream)
{
    const float* hs    = (const float*)d_in[0];   // [S,B,H] f32 -> [T,H]
    const float* probs = (const float*)d_in[1];   // [T,E]  f32 (zeros off top-k)
    // d_in[2] (routing_map) is unused: probs is already masked.
    float* out = (float*)d_out;                   // [T,H] f32

    const int T = in_sizes[1] / NEXP;             // 16384
    const int nblocks = T / TOK_PER_BLK;          // 1024

    moe_combine_wmma_kernel<<<dim3(nblocks), dim3(BLK_THREADS), 0, stream>>>(
        hs, probs, out);
}